// KPToSkl_49692771615129
// MI455X (gfx1250) — compile-verified
//
#include <hip/hip_runtime.h>

// KPToSkl heatmap: out[b,y,x] = max_e exp(-GAMMA * dist2(pixel, segment_e))
//                = exp(-GAMMA * min_e dist2)   (exp monotone)
// dist2 via two rank-2(+affine) matmuls mapped onto V_WMMA_F32_16X16X4_F32.

typedef __attribute__((ext_vector_type(2))) float v2f;
typedef __attribute__((ext_vector_type(8))) float v8f;

#define N_KPS   18
#define N_EPAD  32          // edges padded to 2 WMMA groups of 16
#define GAMMA   0.2f
#define HW      256
#define LOG2E   1.44269504088896340736f

__global__ __launch_bounds__(256) void kps_skel_wmma_kernel(
    const float* __restrict__ kps,   // (32, 18, 2)  last dim = (x, y) pre-reversal
    float* __restrict__ out)         // (32, 256, 256)
{
    // Edge parameter tables (structure-of-arrays), 32 padded edges = 1 KB LDS.
    __shared__ float s_vy[N_EPAD], s_vx[N_EPAD], s_npjv[N_EPAD];
    __shared__ float s_m2py[N_EPAD], s_m2px[N_EPAD], s_pj2[N_EPAD];
    __shared__ float s_invn[N_EPAD], s_vn[N_EPAD];

    const int tid  = threadIdx.x;
    const int lane = tid & 31;
    const int wib  = tid >> 5;
    const int gw   = blockIdx.x * 8 + wib;   // 131072 waves total
    const int b    = gw >> 12;               // 4096 waves (2^12) per batch
    const int row  = (gw >> 4) & 255;
    const int col16= gw & 15;

    // ---- per-block edge params (batch uniform across block) ----
    if (tid < N_EPAD) {
        if (tid == 0) __builtin_prefetch(kps + b * (N_KPS * 2), 0, 0); // global_prefetch_b8
        if (tid < N_KPS) {
            const float* kp = kps + b * (N_KPS * 2);
            const int i = tid, j = (tid + 1) % N_KPS;
            // reference reverses last dim: y = kps[...,1], x = kps[...,0]
            const float piy = kp[i*2+1], pix = kp[i*2+0];
            const float pjy = kp[j*2+1], pjx = kp[j*2+0];
            const float vy = piy - pjy, vx = pix - pjx;
            const float vn = vy*vy + vx*vx;
            s_vy[tid]   = vy;            s_vx[tid]   = vx;
            s_npjv[tid] = -(pjy*vy + pjx*vx);
            s_m2py[tid] = -2.0f * pjy;   s_m2px[tid] = -2.0f * pjx;
            s_pj2[tid]  = pjy*pjy + pjx*pjx;
            s_invn[tid] = 1.0f / vn;     s_vn[tid]   = vn;
        } else {
            // dummy edges: uv=0, rs=0, d2 = |p|^2 + 1e30 -> never the min
            s_vy[tid] = 0.0f;   s_vx[tid] = 0.0f;  s_npjv[tid] = 0.0f;
            s_m2py[tid] = 0.0f; s_m2px[tid] = 0.0f; s_pj2[tid] = 1e30f;
            s_invn[tid] = 0.0f; s_vn[tid] = 0.0f;
        }
    }
    __syncthreads();

    // ---- pixel geometry for this wave's 16-pixel tile ----
    const int  q   = lane & 15;          // pixel index in tile (also A-row idx)
    const bool hi  = lane >= 16;
    const int  col = col16 * 16 + q;
    const float y  = 2.0f * (float)row * (1.0f / 255.0f) - 1.0f;
    const float x  = 2.0f * (float)col * (1.0f / 255.0f) - 1.0f;
    const float p2 = y*y + x*x;

    // B (4x16, K-major): lanes 0-15 hold (K0,K1)=(y,x); lanes 16-31 hold (K2,K3)=(1,0)
    v2f bvec;
    bvec.x = hi ? 1.0f : y;
    bvec.y = hi ? 0.0f : x;

    // A (16x4, M=edge): lanes 0-15 hold (K0,K1); lanes 16-31 hold (K2,K3)
    const int eg0 = q, eg1 = 16 + q;
    v2f a1g0, a2g0, a1g1, a2g1;
    a1g0.x = hi ? s_npjv[eg0] : s_vy[eg0];   a1g0.y = hi ? 0.0f : s_vx[eg0];
    a2g0.x = hi ? s_pj2[eg0]  : s_m2py[eg0]; a2g0.y = hi ? 0.0f : s_m2px[eg0];
    a1g1.x = hi ? s_npjv[eg1] : s_vy[eg1];   a1g1.y = hi ? 0.0f : s_vx[eg1];
    a2g1.x = hi ? s_pj2[eg1]  : s_m2py[eg1]; a2g1.y = hi ? 0.0f : s_m2px[eg1];

    // D[edge][pixel]: uv = p.v - pj.v ; su = -2 p.pj + |pj|^2
    v8f zc = {};
    v8f uv0 = __builtin_amdgcn_wmma_f32_16x16x4_f32(false, a1g0, false, bvec, (short)0, zc, false, false);
    v8f su0 = __builtin_amdgcn_wmma_f32_16x16x4_f32(false, a2g0, false, bvec, (short)0, zc, false, false);
    v8f uv1 = __builtin_amdgcn_wmma_f32_16x16x4_f32(false, a1g1, false, bvec, (short)0, zc, false, false);
    v8f su1 = __builtin_amdgcn_wmma_f32_16x16x4_f32(false, a2g1, false, bvec, (short)0, zc, false, false);

    // ---- per-lane tail: clamp + dist2 + min over this lane's 16 edges ----
    // VGPR r holds edge (r + 8*hi) of each group for pixel q.
    const int ebase = hi ? 8 : 0;
    float md = 1e30f;
#pragma unroll
    for (int r = 0; r < 8; ++r) {
        const int e0 = ebase + r;
        const int e1 = 16 + e0;
        const float u0  = uv0[r];
        const float rs0 = fminf(fmaxf(u0 * s_invn[e0], 0.0f), 1.0f);
        const float d0  = p2 + su0[r] + rs0 * (rs0 * s_vn[e0] - 2.0f * u0);
        const float u1  = uv1[r];
        const float rs1 = fminf(fmaxf(u1 * s_invn[e1], 0.0f), 1.0f);
        const float d1  = p2 + su1[r] + rs1 * (rs1 * s_vn[e1] - 2.0f * u1);
        md = fminf(md, fminf(d0, d1));
    }
    // combine lane l with lane l^16 (other 8+8 edges of the same pixel), wave32
    md = fminf(md, __shfl_xor(md, 16, 32));

    const float heat = __builtin_amdgcn_exp2f(md * (-GAMMA * LOG2E));
    if (!hi) {
        out[((b * HW) + row) * HW + col] = heat;   // 16 consecutive pixels per wave
    }
}

extern "C" void kernel_launch(void* const* d_in, const int* in_sizes, int n_in,
                              void* d_out, int out_size, void* d_ws, size_t ws_size,
                              hipStream_t stream) {
    const float* kps = (const float*)d_in[0];   // (32,18,2) f32
    float* out = (float*)d_out;                 // (32,256,256) f32
    // 2^21 pixels / (16 px per wave) = 131072 waves; 8 waves (256 thr) per block
    dim3 grid(16384), block(256);
    hipLaunchKernelGGL(kps_skel_wmma_kernel, grid, block, 0, stream, kps, out);
    (void)in_sizes; (void)n_in; (void)out_size; (void)d_ws; (void)ws_size;
}